// BallQuery_70153995813273
// MI455X (gfx1250) — compile-verified
//
#include <hip/hip_runtime.h>
#include <float.h>

typedef __attribute__((ext_vector_type(2))) float v2f;
typedef __attribute__((ext_vector_type(8))) float v8f;

#define B_    4
#define N_    8192
#define M_    2048
#define C_    64
#define T_    64
#define K_    32
#define CHUNK 256
#define MT    16      // centers per workgroup (one WMMA M-tile)
#define THREADS 256   // 8 waves
#define NCHUNK (N_ / CHUNK)

// -------------------------------------------------------------------------
// Kernel 1: fused WMMA distance + top-K selection, with async-to-LDS
// double-buffered point staging (ASYNCcnt pipeline).
// dist_rank[m,n] = p2[n] - 2*c[m].p[n]   (c2 term dropped: rank-invariant)
// via V_WMMA_F32_16X16X4_F32 with A[m,:]=(-2cx,-2cy,-2cz,1), B[:,n]=(x,y,z,p2)
// -------------------------------------------------------------------------
__global__ __launch_bounds__(THREADS)
void knn_topk_wmma(const float* __restrict__ pc,   // [B,3,N]
                   const float* __restrict__ cc,   // [B,3,M]
                   int* __restrict__ outIdx)       // [B,M,K]
{
    __shared__ float sPts[2][3][CHUNK];   // double-buffered async staging (6 KB)
    __shared__ float sDist[MT * CHUNK];   // 16 KB distance slab
    __shared__ float hD[MT * K_];         // per-center max-heap (distances)
    __shared__ int   hI[MT * K_];         // per-center max-heap (indices)

    const int wg    = blockIdx.x;               // b * (M/MT) + mTile
    const int b     = wg / (M_ / MT);
    const int mBase = (wg % (M_ / MT)) * MT;
    const int tid   = threadIdx.x;
    const int lane  = tid & 31;
    const int wave  = tid >> 5;
    const int  l15  = lane & 15;
    const bool hiHalf = (lane & 16) != 0;

    for (int i = tid; i < MT * K_; i += THREADS) { hD[i] = FLT_MAX; hI[i] = 0x7fffffff; }

    // A matrix (constant across the whole N loop), per the 32-bit 16x4 layout:
    // VGPR0: lanes0-15 K=0, lanes16-31 K=2 ; VGPR1: lanes0-15 K=1, lanes16-31 K=3
    const float cx = cc[(b*3 + 0)*M_ + mBase + l15];
    const float cy = cc[(b*3 + 1)*M_ + mBase + l15];
    const float cz = cc[(b*3 + 2)*M_ + mBase + l15];
    v2f Amat;
    Amat.x = hiHalf ? (-2.0f * cz) : (-2.0f * cx);
    Amat.y = hiHalf ? 1.0f         : (-2.0f * cy);

    const float* __restrict__ pcb = pc + (size_t)b * 3 * N_;  // batch base

    // Issue one chunk's staging: 256 threads x 3 coords, 1 b32/lane each.
    // GVS addressing: saddr = pcb, per-lane signed 32-bit byte offset.
    // LDS dest byte offset = low 32 bits of generic shared pointer.
    auto issue = [&](int buf, int base) {
        const unsigned goff = (unsigned)((base + tid) * 4);
        const unsigned l0 = (unsigned)(size_t)&sPts[buf][0][tid];
        const unsigned l1 = (unsigned)(size_t)&sPts[buf][1][tid];
        const unsigned l2 = (unsigned)(size_t)&sPts[buf][2][tid];
        asm volatile("global_load_async_to_lds_b32 %0, %1, %2"
                     :: "v"(l0), "v"(goff),               "s"(pcb) : "memory");
        asm volatile("global_load_async_to_lds_b32 %0, %1, %2"
                     :: "v"(l1), "v"(goff + 4u * N_),     "s"(pcb) : "memory");
        asm volatile("global_load_async_to_lds_b32 %0, %1, %2"
                     :: "v"(l2), "v"(goff + 8u * N_),     "s"(pcb) : "memory");
    };

    issue(0, 0);                       // prime the pipeline
    int cur = 0;
    __syncthreads();                   // covers heap init

    for (int chunk = 0; chunk < NCHUNK; ++chunk) {
        const int base = chunk * CHUNK;

        if (chunk + 1 < NCHUNK) {
            issue(cur ^ 1, base + CHUNK);                 // prefetch next chunk
            asm volatile("s_wait_asynccnt 3" ::: "memory"); // current batch done
        } else {
            asm volatile("s_wait_asynccnt 0" ::: "memory");
        }
        __syncthreads();   // whole staging buffer visible to all waves

        // 8 waves x 2 tiles x (16x16) = full 16x256 slab
        #pragma unroll
        for (int tt = 0; tt < 2; ++tt) {
            const int t  = wave * 2 + tt;
            const int nl = t * 16 + l15;
            const float x  = sPts[cur][0][nl];
            const float y  = sPts[cur][1][nl];
            const float z  = sPts[cur][2][nl];
            const float p2 = x*x + y*y + z*z;
            v2f Bmat;
            Bmat.x = hiHalf ? z  : x;     // K=0 / K=2 rows
            Bmat.y = hiHalf ? p2 : y;     // K=1 / K=3 rows
            v8f acc = {};
            acc = __builtin_amdgcn_wmma_f32_16x16x4_f32(
                      false, Amat, false, Bmat, (short)0, acc, false, false);
            // D layout: VGPR j -> (M=j, N=lane) lanes0-15, (M=8+j, N=lane-16)
            const int mloc = hiHalf ? 8 : 0;
            #pragma unroll
            for (int j = 0; j < 8; ++j)
                sDist[(mloc + j) * CHUNK + nl] = acc[j];
        }
        __syncthreads();

        // One thread per center: threshold-rejecting max-heap insert (size 32)
        if (tid < MT) {
            float* __restrict__ h  = hD + tid * K_;
            int*   __restrict__ hx = hI + tid * K_;
            const float* __restrict__ row = sDist + tid * CHUNK;
            for (int n = 0; n < CHUNK; ++n) {
                const float d = row[n];
                if (d < h[0]) {
                    float vd = d; int vi = base + n;
                    int p = 0;
                    while (true) {
                        int cL = 2 * p + 1;
                        if (cL >= K_) break;
                        int cR = cL + 1;
                        int cBig = (cR < K_ && h[cR] > h[cL]) ? cR : cL;
                        if (h[cBig] > vd) { h[p] = h[cBig]; hx[p] = hx[cBig]; p = cBig; }
                        else break;
                    }
                    h[p] = vd; hx[p] = vi;
                }
            }
        }
        __syncthreads();
        cur ^= 1;
    }

    // Sort each center's 32 ascending by (dist, idx) to match argsort order.
    if (tid < MT) {
        float* __restrict__ h  = hD + tid * K_;
        int*   __restrict__ hx = hI + tid * K_;
        for (int i = 1; i < K_; ++i) {
            float vd = h[i]; int vi = hx[i];
            int j = i - 1;
            while (j >= 0 && (h[j] > vd || (h[j] == vd && hx[j] > vi))) {
                h[j + 1] = h[j]; hx[j + 1] = hx[j]; --j;
            }
            h[j + 1] = vd; hx[j + 1] = vi;
        }
        for (int k = 0; k < K_; ++k)
            outIdx[((size_t)b * M_ + mBase + tid) * K_ + k] = hx[k];
    }
}

// -------------------------------------------------------------------------
// Kernel 2: gather. Output = [B,3+C,M,K] (coords rel. to center ++ features)
// followed by [B,T,M,K] temb. Writes coalesced along K; reads are L2-resident
// (whole input set ~17 MB << 192 MB L2).
// -------------------------------------------------------------------------
__global__ __launch_bounds__(256)
void gather_out(const float* __restrict__ pc,
                const float* __restrict__ cc,
                const float* __restrict__ temb,
                const float* __restrict__ pf,
                const int*   __restrict__ idxBuf,
                float*       __restrict__ out)
{
    const long long total1 = (long long)B_ * (3 + C_) * M_ * K_;
    const long long total2 = (long long)B_ * T_ * M_ * K_;
    long long i = (long long)blockIdx.x * blockDim.x + threadIdx.x;
    if (i >= total1 + total2) return;

    if (i < total1) {
        int k = (int)(i % K_);  long long r = i / K_;
        int m = (int)(r % M_);  r /= M_;
        int c = (int)(r % (3 + C_));
        int b = (int)(r / (3 + C_));
        int id = idxBuf[((long long)b * M_ + m) * K_ + k];
        float v;
        if (c < 3)
            v = pc[((long long)b * 3 + c) * N_ + id] - cc[((long long)b * 3 + c) * M_ + m];
        else
            v = pf[((long long)b * C_ + (c - 3)) * N_ + id];
        out[i] = v;
    } else {
        long long j = i - total1;
        int k = (int)(j % K_);  long long r = j / K_;
        int m = (int)(r % M_);  r /= M_;
        int t = (int)(r % T_);
        int b = (int)(r / T_);
        int id = idxBuf[((long long)b * M_ + m) * K_ + k];
        out[i] = temb[((long long)b * T_ + t) * N_ + id];
    }
}

extern "C" void kernel_launch(void* const* d_in, const int* in_sizes, int n_in,
                              void* d_out, int out_size, void* d_ws, size_t ws_size,
                              hipStream_t stream) {
    const float* pc   = (const float*)d_in[0];   // [B,3,N]
    const float* cc   = (const float*)d_in[1];   // [B,3,M]
    const float* temb = (const float*)d_in[2];   // [B,T,N]
    const float* pf   = (const float*)d_in[3];   // [B,C,N]
    float* out = (float*)d_out;
    int*   idxBuf = (int*)d_ws;                  // B*M*K ints = 1 MB

    // Phase 1: WMMA distances + async-pipelined top-K selection
    const int nBlocks1 = B_ * (M_ / MT);         // 512
    knn_topk_wmma<<<nBlocks1, THREADS, 0, stream>>>(pc, cc, idxBuf);

    // Phase 2: gather into concatenated output
    const long long total = (long long)B_ * (3 + C_) * M_ * K_
                          + (long long)B_ * T_ * M_ * K_;
    const int nBlocks2 = (int)((total + 255) / 256);
    gather_out<<<nBlocks2, 256, 0, stream>>>(pc, cc, temb, pf, idxBuf, out);
}